// MinimalGN_55688545960168
// MI455X (gfx1250) — compile-verified
//
#include <hip/hip_runtime.h>

typedef __attribute__((ext_vector_type(2))) float v2f;
typedef __attribute__((ext_vector_type(8))) float v8f;

#define N_NODES 50000
#define N_EDGES 1600000
#define D_IN    128
#define D_EDGE  64
#define D_NODE  128
#define M_TILES (N_NODES / 16)   // 3125 exact

// ---------------------------------------------------------------------------
// Kernel 0: zero the agg buffer (int view; 0 == 0.0f bit pattern).
// ---------------------------------------------------------------------------
__global__ void __launch_bounds__(256) zero_agg_kernel(int4* __restrict__ agg4) {
    int i = blockIdx.x * blockDim.x + threadIdx.x;
    if (i < N_NODES * D_EDGE / 4) {
        agg4[i] = make_int4(0, 0, 0, 0);
    }
}

// ---------------------------------------------------------------------------
// Kernel 1: proj[50000x64] = NF[50000x128] @ Wfs[128x64] + bfs
// One wave computes a 16x64 output stripe using V_WMMA_F32_16X16X4_F32.
// A 16x4 f32 layout: lanes 0-15 -> (M=lane, K=k,k+1), lanes 16-31 -> K=k+2,k+3.
// B 4x16 f32 layout: VGPR0/1 hold rows K=kb,kb+1 striped across 16 lanes.
// C/D: VGPR r -> row (r + 8*half), col = lane&15.
// ---------------------------------------------------------------------------
__global__ void __launch_bounds__(256) gemm_fs_kernel(
        const float* __restrict__ NF, const float* __restrict__ Wfs,
        const float* __restrict__ bfs, float* __restrict__ proj) {
    int wave = blockIdx.x * (blockDim.x >> 5) + (threadIdx.x >> 5);
    if (wave >= M_TILES) return;                 // wave-uniform: EXEC stays full
    int lane = threadIdx.x & 31;
    int half = lane >> 4;
    int lm   = lane & 15;
    int m0   = wave * 16;

    const float* Arow = NF + (size_t)(m0 + lm) * D_IN;

    v8f c0 = {}, c1 = {}, c2 = {}, c3 = {};
    for (int k = 0; k < D_IN; k += 4) {
        int kb = k + 2 * half;
        v2f a = *(const v2f*)(Arow + kb);        // 8B aligned (kb even)
        const float* Bp = Wfs + (size_t)kb * D_EDGE + lm;
        v2f b0, b1, b2, b3;
        b0.x = Bp[0];  b0.y = Bp[D_EDGE + 0];
        b1.x = Bp[16]; b1.y = Bp[D_EDGE + 16];
        b2.x = Bp[32]; b2.y = Bp[D_EDGE + 32];
        b3.x = Bp[48]; b3.y = Bp[D_EDGE + 48];
        c0 = __builtin_amdgcn_wmma_f32_16x16x4_f32(false, a, false, b0, (short)0, c0, false, false);
        c1 = __builtin_amdgcn_wmma_f32_16x16x4_f32(false, a, false, b1, (short)0, c1, false, false);
        c2 = __builtin_amdgcn_wmma_f32_16x16x4_f32(false, a, false, b2, (short)0, c2, false, false);
        c3 = __builtin_amdgcn_wmma_f32_16x16x4_f32(false, a, false, b3, (short)0, c3, false, false);
    }

    float bias0 = bfs[lm + 0];
    float bias1 = bfs[lm + 16];
    float bias2 = bfs[lm + 32];
    float bias3 = bfs[lm + 48];
#pragma unroll
    for (int r = 0; r < 8; ++r) {
        int row = m0 + r + 8 * half;
        float* o = proj + (size_t)row * D_EDGE + lm;
        o[0]  = c0[r] + bias0;
        o[16] = c1[r] + bias1;
        o[32] = c2[r] + bias2;
        o[48] = c3[r] + bias3;
    }
}

// ---------------------------------------------------------------------------
// Kernel 2: scatter-max over edges.
// agg zero-init + signed-int atomicMax on float bits == segment_max(relu(.))
// with empty-segment->0 semantics (negative floats are negative ints -> lose
// to 0; positive floats order identically to their int bit patterns).
// 16 threads per edge, float4 gather (L2-resident proj), 4 atomics each.
// ---------------------------------------------------------------------------
__global__ void __launch_bounds__(256) scatter_max_kernel(
        const float* __restrict__ proj, const int* __restrict__ snd,
        const int* __restrict__ rcv, int* __restrict__ agg) {
    unsigned tid = blockIdx.x * 256u + threadIdx.x;
    if (tid >= (unsigned)N_EDGES * 16u) return;
    int e = tid >> 4;
    int k = (tid & 15) << 2;
    int s = snd[e];
    int r = rcv[e];
    float4 v = *(const float4*)(proj + (size_t)s * D_EDGE + k);
    int* dst = agg + (size_t)r * D_EDGE + k;
    atomicMax(dst + 0, __float_as_int(v.x));
    atomicMax(dst + 1, __float_as_int(v.y));
    atomicMax(dst + 2, __float_as_int(v.z));
    atomicMax(dst + 3, __float_as_int(v.w));
}

// ---------------------------------------------------------------------------
// Kernel 3: out[50000x128] = NF @ Wgn + agg @ Wgin + (bgn + bgin)
// One wave computes a 16x128 stripe: 8 WMMA accumulators, concatenated K loop
// (128 over NF/Wgn, then 64 over agg/Wgin).
// ---------------------------------------------------------------------------
__global__ void __launch_bounds__(256) gemm_out_kernel(
        const float* __restrict__ NF, const float* __restrict__ agg,
        const float* __restrict__ Wgn, const float* __restrict__ Wgin,
        const float* __restrict__ bgn, const float* __restrict__ bgin,
        float* __restrict__ out) {
    int wave = blockIdx.x * (blockDim.x >> 5) + (threadIdx.x >> 5);
    if (wave >= M_TILES) return;
    int lane = threadIdx.x & 31;
    int half = lane >> 4;
    int lm   = lane & 15;
    int m0   = wave * 16;

    v8f c[8];
#pragma unroll
    for (int nt = 0; nt < 8; ++nt) c[nt] = (v8f){};

    // --- K segment 1: NF (K = 128) @ Wgn ---
    {
        const float* Arow = NF + (size_t)(m0 + lm) * D_IN;
        for (int k = 0; k < D_IN; k += 4) {
            int kb = k + 2 * half;
            v2f a = *(const v2f*)(Arow + kb);
            const float* Bp = Wgn + (size_t)kb * D_NODE + lm;
#pragma unroll
            for (int nt = 0; nt < 8; ++nt) {
                v2f b;
                b.x = Bp[nt * 16];
                b.y = Bp[D_NODE + nt * 16];
                c[nt] = __builtin_amdgcn_wmma_f32_16x16x4_f32(false, a, false, b,
                                                              (short)0, c[nt], false, false);
            }
        }
    }
    // --- K segment 2: agg (K = 64) @ Wgin ---
    {
        const float* Arow = agg + (size_t)(m0 + lm) * D_EDGE;
        for (int k = 0; k < D_EDGE; k += 4) {
            int kb = k + 2 * half;
            v2f a = *(const v2f*)(Arow + kb);
            const float* Bp = Wgin + (size_t)kb * D_NODE + lm;
#pragma unroll
            for (int nt = 0; nt < 8; ++nt) {
                v2f b;
                b.x = Bp[nt * 16];
                b.y = Bp[D_NODE + nt * 16];
                c[nt] = __builtin_amdgcn_wmma_f32_16x16x4_f32(false, a, false, b,
                                                              (short)0, c[nt], false, false);
            }
        }
    }

#pragma unroll
    for (int nt = 0; nt < 8; ++nt) {
        int n = nt * 16 + lm;
        float bias = bgn[n] + bgin[n];
#pragma unroll
        for (int r = 0; r < 8; ++r) {
            int row = m0 + r + 8 * half;
            out[(size_t)row * D_NODE + n] = c[nt][r] + bias;
        }
    }
}

// ---------------------------------------------------------------------------
extern "C" void kernel_launch(void* const* d_in, const int* in_sizes, int n_in,
                              void* d_out, int out_size, void* d_ws, size_t ws_size,
                              hipStream_t stream) {
    (void)in_sizes; (void)n_in; (void)out_size; (void)ws_size;
    const float* NF   = (const float*)d_in[0];
    const float* Wfs  = (const float*)d_in[1];
    const float* bfs  = (const float*)d_in[2];
    const float* Wgn  = (const float*)d_in[3];
    const float* bgn  = (const float*)d_in[4];
    const float* Wgin = (const float*)d_in[5];
    const float* bgin = (const float*)d_in[6];
    const int*   snd  = (const int*)d_in[7];
    const int*   rcv  = (const int*)d_in[8];
    float* out = (float*)d_out;

    float* proj = (float*)d_ws;                                       // 12.8 MB
    int*   agg  = (int*)((char*)d_ws + (size_t)N_NODES * D_EDGE * 4); // 12.8 MB

    // 0) zero agg (fresh each call; harness does not re-poison between replays)
    {
        int n4 = N_NODES * D_EDGE / 4;
        zero_agg_kernel<<<(n4 + 255) / 256, 256, 0, stream>>>((int4*)agg);
    }
    // 1) proj = NF @ Wfs + bfs
    {
        int threads = M_TILES * 32;
        gemm_fs_kernel<<<(threads + 255) / 256, 256, 0, stream>>>(NF, Wfs, bfs, proj);
    }
    // 2) scatter-max over edges
    {
        unsigned total = (unsigned)N_EDGES * 16u;
        scatter_max_kernel<<<(total + 255u) / 256u, 256, 0, stream>>>(proj, snd, rcv, agg);
    }
    // 3) out = NF @ Wgn + agg @ Wgin + biases
    {
        int threads = M_TILES * 32;
        gemm_out_kernel<<<(threads + 255) / 256, 256, 0, stream>>>(
            NF, (const float*)agg, Wgn, Wgin, bgn, bgin, out);
    }
}